// BaseRGCN_3195455668259
// MI455X (gfx1250) — compile-verified
//
#include <hip/hip_runtime.h>
#include <hip/hip_bf16.h>

typedef float v2f __attribute__((ext_vector_type(2)));
typedef float v8f __attribute__((ext_vector_type(8)));

#define N_NODES 100000
#define N_EDGES 1600000
#define N_REL   8
#define IN_CH   128
#define HID     64
#define OUT_CH  128

// ---------------------------------------------------------------------------
// C[M,F] = A[M,K] @ B[K,F] (+ bias broadcast over rows, if bias != nullptr)
// One wave32 per 16-row strip; F/16 accumulator tiles; K-loop step 4 using
// V_WMMA_F32_16X16X4_F32. B (<= 32KB) staged in LDS.
// ---------------------------------------------------------------------------
template<int K, int F>
__global__ __launch_bounds__(256)
void gemm_wmma_f32(const float* __restrict__ A, const float* __restrict__ B,
                   const float* __restrict__ bias, float* __restrict__ C, int M)
{
    constexpr int NT = F / 16;
    __shared__ float Bs[K * F];
    for (int i = threadIdx.x; i < K * F; i += 256) Bs[i] = B[i];
    __syncthreads();

    const int wave = blockIdx.x * 8 + (threadIdx.x >> 5);
    const int lane = threadIdx.x & 31;
    const int m0   = wave * 16;
    if (m0 >= M) return;                 // wave-uniform: EXEC stays all-ones

    const int laneLo = lane & 15;        // row (A) / col (B,C)
    const int laneHi = lane >> 4;        // selects K-half / M-half

    v8f zero = {0.f, 0.f, 0.f, 0.f, 0.f, 0.f, 0.f, 0.f};
    v8f acc[NT];
#pragma unroll
    for (int t = 0; t < NT; ++t) acc[t] = zero;

    const float* Arow = A + (size_t)(m0 + laneLo) * K;
    for (int k0 = 0; k0 < K; k0 += 4) {
        // A 16x4 fragment: lanes 0-15 hold K={k0,k0+1}, lanes 16-31 K={k0+2,k0+3}
        v2f a;
        a.x = Arow[k0 + 2 * laneHi + 0];
        a.y = Arow[k0 + 2 * laneHi + 1];
#pragma unroll
        for (int t = 0; t < NT; ++t) {
            // B 4x16 fragment: VGPR v holds row k0 + 2*laneHi + v, col = laneLo
            const float* bp = &Bs[(k0 + 2 * laneHi) * F + t * 16 + laneLo];
            v2f b;
            b.x = bp[0];
            b.y = bp[F];
            acc[t] = __builtin_amdgcn_wmma_f32_16x16x4_f32(
                false, a, false, b, (short)0, acc[t], false, false);
        }
    }

    // C/D layout: lanes 0-15 -> rows m0+0..7, lanes 16-31 -> rows m0+8..15
    const int rowBase = m0 + laneHi * 8;
#pragma unroll
    for (int t = 0; t < NT; ++t) {
        const int col = t * 16 + laneLo;
        const float bv = bias ? bias[col] : 0.0f;
#pragma unroll
        for (int i = 0; i < 8; ++i)
            C[(size_t)(rowBase + i) * F + col] = acc[t][i] + bv;
    }
}

// ---------------------------------------------------------------------------
// Per-(dst,rel) degree counting: deg[dst*8 + et] += 1
// ---------------------------------------------------------------------------
__global__ void zero_f32(float* __restrict__ p, int n)
{
    int i = blockIdx.x * blockDim.x + threadIdx.x;
    if (i < n) p[i] = 0.0f;
}

__global__ void deg_count(const int* __restrict__ dst, const int* __restrict__ et,
                          float* __restrict__ deg, int E)
{
    int e = blockIdx.x * blockDim.x + threadIdx.x;
    if (e < E) atomicAdd(&deg[(size_t)dst[e] * N_REL + et[e]], 1.0f);
}

// ---------------------------------------------------------------------------
// For edges with et==rel: out[dst] += tmp[src] / max(deg[dst*8+rel],1)
// One wave per edge; lanes stride the F features (wave-uniform branches only).
// ---------------------------------------------------------------------------
template<int F>
__global__ __launch_bounds__(256)
void edge_scatter(const int* __restrict__ src, const int* __restrict__ dst,
                  const int* __restrict__ et, const float* __restrict__ deg,
                  const float* __restrict__ tmp, float* __restrict__ out,
                  int E, int rel)
{
    int e = (blockIdx.x * blockDim.x + threadIdx.x) >> 5;
    int lane = threadIdx.x & 31;
    if (e >= E) return;
    int r = et[e];
    if (r != rel) return;
    int s = src[e];
    int d = dst[e];
    float norm = 1.0f / fmaxf(deg[(size_t)d * N_REL + r], 1.0f);
    const float* ts = tmp + (size_t)s * F;
    float* od = out + (size_t)d * F;
#pragma unroll
    for (int f = lane; f < F; f += 32)
        atomicAdd(&od[f], ts[f] * norm);
}

__global__ void relu_f32(float* __restrict__ p, int n)
{
    int i = blockIdx.x * blockDim.x + threadIdx.x;
    if (i < n) p[i] = fmaxf(p[i], 0.0f);
}

// ---------------------------------------------------------------------------
extern "C" void kernel_launch(void* const* d_in, const int* in_sizes, int n_in,
                              void* d_out, int out_size, void* d_ws, size_t ws_size,
                              hipStream_t stream)
{
    const float* x     = (const float*)d_in[0];
    const int*   eidx  = (const int*)  d_in[1];   // [2, E]
    const int*   etype = (const int*)  d_in[2];   // [E]
    const float* W1    = (const float*)d_in[3];   // [R, 128, 64]
    const float* root1 = (const float*)d_in[4];   // [128, 64]
    const float* b1    = (const float*)d_in[5];   // [64]
    const float* W2    = (const float*)d_in[6];   // [R, 64, 128]
    const float* root2 = (const float*)d_in[7];   // [64, 128]
    const float* b2    = (const float*)d_in[8];   // [128]
    float* out = (float*)d_out;                   // [N, 128]

    const int* src = eidx;
    const int* dst = eidx + N_EDGES;

    // Workspace layout (bytes): deg @ 0 (3.2MB), tmp @ 4MB (<=51.2MB), h @ 60MB (25.6MB)
    char* ws  = (char*)d_ws;
    float* deg = (float*)(ws);
    float* tmp = (float*)(ws + (size_t)4  * 1024 * 1024);
    float* h   = (float*)(ws + (size_t)60 * 1024 * 1024);

    const int degN      = N_NODES * N_REL;
    const int gemmBlks  = (N_NODES / 16 + 7) / 8;          // 6250 waves / 8 per block
    const int edgeBlks  = (N_EDGES + 7) / 8;               // one wave per edge
    const int eltBlks   = (N_EDGES + 255) / 256;

    // ---- degrees (shared by both layers; graph is static within the call) ----
    zero_f32<<<(degN + 255) / 256, 256, 0, stream>>>(deg, degN);
    deg_count<<<eltBlks, 256, 0, stream>>>(dst, etype, deg, N_EDGES);

    // ---- layer 1: h = relu( sum_r scatter(x@W1[r]) + x@root1 + b1 ) ----
    gemm_wmma_f32<IN_CH, HID><<<gemmBlks, 256, 0, stream>>>(x, root1, b1, h, N_NODES);
    for (int r = 0; r < N_REL; ++r) {
        gemm_wmma_f32<IN_CH, HID><<<gemmBlks, 256, 0, stream>>>(
            x, W1 + (size_t)r * IN_CH * HID, nullptr, tmp, N_NODES);
        edge_scatter<HID><<<edgeBlks, 256, 0, stream>>>(
            src, dst, etype, deg, tmp, h, N_EDGES, r);
    }
    relu_f32<<<(N_NODES * HID + 255) / 256, 256, 0, stream>>>(h, N_NODES * HID);

    // ---- layer 2: out = sum_r scatter(h@W2[r]) + h@root2 + b2 ----
    gemm_wmma_f32<HID, OUT_CH><<<gemmBlks, 256, 0, stream>>>(h, root2, b2, out, N_NODES);
    for (int r = 0; r < N_REL; ++r) {
        gemm_wmma_f32<HID, OUT_CH><<<gemmBlks, 256, 0, stream>>>(
            h, W2 + (size_t)r * HID * OUT_CH, nullptr, tmp, N_NODES);
        edge_scatter<OUT_CH><<<edgeBlks, 256, 0, stream>>>(
            src, dst, etype, deg, tmp, out, N_EDGES, r);
    }
}